// EnsemblePrompt_35708358099008
// MI455X (gfx1250) — compile-verified
//
#include <hip/hip_runtime.h>

#define B_      1024
#define D_      768
#define E_      4096
#define P_      1024
#define L_      8
#define S_      8
#define POOLS_  4
#define NA_     204
#define EPS_    1e-8f

typedef float v2f __attribute__((ext_vector_type(2)));
typedef float v8f __attribute__((ext_vector_type(8)));

// ---------------------------------------------------------------------------
// GEMM via V_WMMA_F32_16X16X4_F32.  C[M,N] = A[M,K] * B, optional (1 - x).
//  B_NT == false : Bm is [K,N] row-major (B used directly)
//  B_NT == true  : Bm is [N,K] row-major (B = Bm^T)
// Each wave computes a 16x64 output strip (4 accumulators, A fragment reused).
// Task count divides grid exactly -> no divergence, EXEC all-ones for WMMA.
// ---------------------------------------------------------------------------
template <bool B_NT, bool ONE_MINUS>
__global__ __launch_bounds__(256) void gemm_wmma_f32(
    const float* __restrict__ A, const float* __restrict__ Bm,
    float* __restrict__ C, int M, int N, int K)
{
    const int lane = threadIdx.x & 31;
    const int wave = blockIdx.x * (blockDim.x >> 5) + (threadIdx.x >> 5);
    const int nsup = N >> 6;                 // number of 64-wide supertiles
    const int mt   = wave / nsup;
    const int ns   = wave % nsup;
    const int row0 = mt << 4;
    const int col0 = ns << 6;

    const int m  = lane & 15;                // A row part / B col part (ISA layout)
    const int kk = (lane >> 4) << 1;         // lanes 0-15 -> K 0,1 ; lanes 16-31 -> K 2,3

    v8f acc[4] = {v8f{}, v8f{}, v8f{}, v8f{}};
    const float* Arow = A + (size_t)(row0 + m) * K;

    for (int k = 0; k < K; k += 4) {
        v2f a;
        a.x = Arow[k + kk];
        a.y = Arow[k + kk + 1];
        v2f b[4];
        if (!B_NT) {
            const float* Bk0 = Bm + (size_t)(k + kk)     * N + col0 + m;
            const float* Bk1 = Bm + (size_t)(k + kk + 1) * N + col0 + m;
#pragma unroll
            for (int t = 0; t < 4; ++t) { b[t].x = Bk0[t * 16]; b[t].y = Bk1[t * 16]; }
        } else {
#pragma unroll
            for (int t = 0; t < 4; ++t) {
                const float* Bn = Bm + (size_t)(col0 + t * 16 + m) * K + k + kk;
                b[t].x = Bn[0]; b[t].y = Bn[1];
            }
        }
#pragma unroll
        for (int t = 0; t < 4; ++t) {
            acc[t] = __builtin_amdgcn_wmma_f32_16x16x4_f32(
                false, a, false, b[t], (short)0, acc[t], false, false);
        }
    }

    const int rbase = row0 + ((lane >> 4) << 3);  // lanes 0-15: rows r ; 16-31: rows r+8
    const int nc    = lane & 15;
#pragma unroll
    for (int t = 0; t < 4; ++t) {
#pragma unroll
        for (int r = 0; r < 8; ++r) {
            float v = acc[t][r];
            if (ONE_MINUS) v = 1.0f - v;
            C[(size_t)(rbase + r) * N + col0 + t * 16 + nc] = v;
        }
    }
}

// ---------------------------------------------------------------------------
// Per-row WTA: keep top NA_ of E_ values (4-pass radix select on monotonic
// float keys), then L2-normalize the surviving entries in place (rest -> 0).
// One 256-thread block per row; each thread caches 16 elements in registers.
// ---------------------------------------------------------------------------
__global__ __launch_bounds__(256) void wta_sparse_normalize(float* __restrict__ sc)
{
    const int row = blockIdx.x;
    const int tid = threadIdx.x;
    float* rowp = sc + (size_t)row * E_;

    float    v[16];
    unsigned key[16];
#pragma unroll
    for (int j = 0; j < 16; ++j) {
        float f = rowp[tid + j * 256];
        v[j] = f;
        unsigned u = __float_as_uint(f);
        key[j] = (u & 0x80000000u) ? ~u : (u | 0x80000000u);  // ascending order key
    }

    __shared__ unsigned hist[256];
    __shared__ unsigned s_prefix, s_remaining;
    __shared__ float    red[256];
    if (tid == 0) { s_prefix = 0u; s_remaining = NA_; }

    unsigned prefix = 0u, mask = 0u;
    for (int pass = 3; pass >= 0; --pass) {
        __syncthreads();
        hist[tid] = 0u;
        __syncthreads();
#pragma unroll
        for (int j = 0; j < 16; ++j)
            if ((key[j] & mask) == prefix)
                atomicAdd(&hist[(key[j] >> (8 * pass)) & 255u], 1u);
        __syncthreads();
        if (tid == 0) {
            unsigned rem = s_remaining;
            unsigned c = 0; int b = 255;
            for (; b > 0; --b) {
                unsigned h = hist[b];
                if (c + h >= rem) break;
                c += h;
            }
            s_remaining = rem - c;
            s_prefix    = s_prefix | ((unsigned)b << (8 * pass));
        }
        __syncthreads();
        prefix = s_prefix;
        mask   = 0xFFFFFFFFu << (8 * pass);
    }
    // prefix == key of the NA_-th largest element; select key >= prefix.
    float ss = 0.f;
#pragma unroll
    for (int j = 0; j < 16; ++j)
        if (key[j] >= prefix) ss += v[j] * v[j];
    red[tid] = ss;
    __syncthreads();
    for (int off = 128; off > 0; off >>= 1) {
        if (tid < off) red[tid] += red[tid + off];
        __syncthreads();
    }
    const float inv = 1.0f / fmaxf(sqrtf(red[0]), EPS_);
#pragma unroll
    for (int j = 0; j < 16; ++j)
        rowp[tid + j * 256] = (key[j] >= prefix) ? v[j] * inv : 0.0f;
}

// ---------------------------------------------------------------------------
// Row-wise L2 normalization of key_bank (clamped at EPS like torch cosine).
// ---------------------------------------------------------------------------
__global__ __launch_bounds__(256) void normalize_rows(
    const float* __restrict__ in, float* __restrict__ out)
{
    const int row = blockIdx.x;
    const int tid = threadIdx.x;
    const float* rp = in  + (size_t)row * E_;
    float*       op = out + (size_t)row * E_;

    float v[16]; float ss = 0.f;
#pragma unroll
    for (int j = 0; j < 16; ++j) { v[j] = rp[tid + j * 256]; ss += v[j] * v[j]; }

    __shared__ float red[256];
    red[tid] = ss;
    __syncthreads();
    for (int off = 128; off > 0; off >>= 1) {
        if (tid < off) red[tid] += red[tid + off];
        __syncthreads();
    }
    const float inv = 1.0f / fmaxf(sqrtf(red[0]), EPS_);
#pragma unroll
    for (int j = 0; j < 16; ++j) op[tid + j * 256] = v[j] * inv;
}

// ---------------------------------------------------------------------------
// Per-row 8 smallest values (+indices) of match[row, 0..P_).  Values emerge
// in ascending order == reference -top_k(-match).
// ---------------------------------------------------------------------------
__global__ __launch_bounds__(256) void top8_min(
    const float* __restrict__ match, float* __restrict__ sim, int* __restrict__ selidx)
{
    const int row = blockIdx.x;
    const int tid = threadIdx.x;
    const float* rp = match + (size_t)row * P_;

    float v[4];
#pragma unroll
    for (int j = 0; j < 4; ++j) v[j] = rp[tid + j * 256];

    __shared__ float rv[256];
    __shared__ int   ri[256];

    for (int s = 0; s < S_; ++s) {
        float best = v[0]; int bi = tid;
#pragma unroll
        for (int j = 1; j < 4; ++j) {
            int idx = tid + j * 256;
            if (v[j] < best || (v[j] == best && idx < bi)) { best = v[j]; bi = idx; }
        }
        rv[tid] = best; ri[tid] = bi;
        __syncthreads();
        for (int off = 128; off > 0; off >>= 1) {
            if (tid < off) {
                if (rv[tid + off] < rv[tid] ||
                    (rv[tid + off] == rv[tid] && ri[tid + off] < ri[tid])) {
                    rv[tid] = rv[tid + off]; ri[tid] = ri[tid + off];
                }
            }
            __syncthreads();
        }
        const int   widx = ri[0];
        const float wval = rv[0];
        if (tid == 0) { sim[row * S_ + s] = wval; selidx[row * S_ + s] = widx; }
        if ((widx & 255) == tid) v[widx >> 8] = __builtin_inff();  // knock out winner
        __syncthreads();
    }
}

// ---------------------------------------------------------------------------
// Gather: selected[pool][b][s*L..][*] = prompts[pool][sel_idx[b][s]][*][*]
// One block per (pool,b,s): 24 KB float4 streaming copy. prompts (96 MB) fits
// in the 192 MB L2 -> reads are mostly cache hits; writes stream to HBM.
// ---------------------------------------------------------------------------
__global__ __launch_bounds__(256) void gather_prompts(
    const float* __restrict__ prompts, const int* __restrict__ selidx,
    float* __restrict__ outSel)
{
    const int task = blockIdx.x;               // pool*B*S + b*S + s
    const int s    = task & (S_ - 1);
    const int b    = (task >> 3) & (B_ - 1);
    const int pool = task >> 13;
    const int p    = selidx[b * S_ + s];

    const float4* src = (const float4*)(prompts +
        (((size_t)pool * P_ + p) * L_) * D_);
    float4* dst = (float4*)(outSel +
        (((size_t)pool * B_ + b) * (S_ * L_) + (size_t)s * L_) * D_);

    const int n4 = (L_ * D_) / 4;              // 1536 float4 per tile
    for (int i = threadIdx.x; i < n4; i += 256) dst[i] = src[i];
}

// ---------------------------------------------------------------------------
extern "C" void kernel_launch(void* const* d_in, const int* in_sizes, int n_in,
                              void* d_out, int out_size, void* d_ws, size_t ws_size,
                              hipStream_t stream)
{
    (void)in_sizes; (void)n_in; (void)out_size; (void)ws_size;
    const float* query   = (const float*)d_in[0];   // [B,D]
    const float* rproj   = (const float*)d_in[1];   // [D,E]
    const float* kbank   = (const float*)d_in[2];   // [P,E]
    const float* prompts = (const float*)d_in[3];   // [POOLS,P,L,D]

    float* out    = (float*)d_out;
    float* sim    = out;                            // [B,S]
    float* outSel = out + (size_t)B_ * S_;          // [POOLS,B,S*L,D]

    float* score  = (float*)d_ws;                   // B*E (in-place -> sn)
    float* kn     = score + (size_t)B_ * E_;        // P*E
    float* match  = kn    + (size_t)P_ * E_;        // B*P
    int*   selidx = (int*)(match + (size_t)B_ * P_);// B*S

    // 1) score = query @ rproj              (wmma f32, 4096 wave-tasks)
    gemm_wmma_f32<false, false><<<(B_ / 16) * (E_ / 64) / 8, 256, 0, stream>>>(
        query, rproj, score, B_, E_, D_);

    // 2) kn = row-normalized key_bank
    normalize_rows<<<P_, 256, 0, stream>>>(kbank, kn);

    // 3) WTA top-204 sparsify + row normalize, in place -> sn
    wta_sparse_normalize<<<B_, 256, 0, stream>>>(score);

    // 4) match = 1 - sn @ kn^T              (wmma f32, 1024 wave-tasks)
    gemm_wmma_f32<true, true><<<(B_ / 16) * (P_ / 64) / 8, 256, 0, stream>>>(
        score, kn, match, B_, P_, E_);

    // 5) per-row 8 smallest distances -> similarity + indices
    top8_min<<<B_, 256, 0, stream>>>(match, sim, selidx);

    // 6) gather selected prompts (memory-bound tail, ~0.9 GB HBM traffic)
    gather_prompts<<<POOLS_ * B_ * S_, 256, 0, stream>>>(prompts, selidx, outSel);
}